// GCNRegression_50139448213625
// MI455X (gfx1250) — compile-verified
//
#include <hip/hip_runtime.h>
#include <hip/hip_bf16.h>

typedef __attribute__((ext_vector_type(2))) float v2f;
typedef __attribute__((ext_vector_type(8))) float v8f;

#define N_NODES  100000
#define N_EDGES  1600000
#define N_GRAPHS 128

// ---------------------------------------------------------------------------
// Generic zero
// ---------------------------------------------------------------------------
__global__ void zero_kernel(float* __restrict__ p, int n) {
    int i = blockIdx.x * blockDim.x + threadIdx.x;
    if (i < n) p[i] = 0.0f;
}

// ---------------------------------------------------------------------------
// Degree accumulation: deg[col[e]] += 1   (hardware f32 atomic add)
// ---------------------------------------------------------------------------
__global__ void degree_kernel(const int* __restrict__ col, float* __restrict__ deg, int E) {
    int e = blockIdx.x * blockDim.x + threadIdx.x;
    if (e < E) unsafeAtomicAdd(&deg[col[e]], 1.0f);
}

// deg -> dinv in place: deg>0 ? deg^-1/2 : 0
__global__ void dinv_kernel(float* __restrict__ deg, int n) {
    int i = blockIdx.x * blockDim.x + threadIdx.x;
    if (i < n) {
        float d = deg[i];
        deg[i] = (d > 0.0f) ? rsqrtf(d) : 0.0f;
    }
}

// ---------------------------------------------------------------------------
// Dense transform H = X @ W  via V_WMMA_F32_16X16X4_F32.
//   X: [N, Cin] row-major, W: [Cin, 64] row-major, H: [N, 64].
//   Block = 128 threads = 4 waves; wave w computes 16x16 tile (rows of the
//   block's row-tile) x (cols 16w..16w+15). N is a multiple of 16 -> EXEC all 1s.
// A 16x4 layout: lane<16 holds K=k,k+1 ; lane>=16 holds K=k+2,k+3 (2 VGPRs).
// B 4x16 layout mirrored. C/D: VGPR r -> row r + 8*(lane>=16), col lane&15.
// ---------------------------------------------------------------------------
__global__ __launch_bounds__(128) void gemm_wmma_f32(
    const float* __restrict__ X, const float* __restrict__ W,
    float* __restrict__ H, int N, int Cin)
{
    const int lane  = threadIdx.x & 31;
    const int wave  = threadIdx.x >> 5;          // 0..3 : column tile
    const int row0  = blockIdx.x * 16;
    const int col0  = wave * 16;

    const int mrow  = lane & 15;                 // M index for A
    const int khalf = (lane >> 4) * 2;           // 0 or 2 : K sub-offset
    const int ncol  = lane & 15;                 // N index for B/C/D

    const float* xrow = X + (size_t)(row0 + mrow) * Cin + khalf;

    v8f acc = {};
    for (int k = 0; k < Cin; k += 4) {
        v2f a;
        a.x = xrow[k];
        a.y = xrow[k + 1];
        v2f b;
        b.x = W[(size_t)(k + khalf)     * 64 + col0 + ncol];
        b.y = W[(size_t)(k + khalf + 1) * 64 + col0 + ncol];
        acc = __builtin_amdgcn_wmma_f32_16x16x4_f32(
            /*neg_a=*/false, a, /*neg_b=*/false, b,
            /*c_mod=*/(short)0, acc, /*reuse_a=*/false, /*reuse_b=*/false);
    }

    const int rbase = (lane >> 4) * 8;           // rows 0..7 or 8..15
#pragma unroll
    for (int r = 0; r < 8; ++r) {
        H[(size_t)(row0 + rbase + r) * 64 + col0 + ncol] = acc[r];
    }
}

// ---------------------------------------------------------------------------
// acc[n][c] = bias[c]
// ---------------------------------------------------------------------------
__global__ void init_bias_kernel(float* __restrict__ acc, const float* __restrict__ b, int total) {
    int i = blockIdx.x * blockDim.x + threadIdx.x;
    if (i < total) acc[i] = b[i & 63];
}

// ---------------------------------------------------------------------------
// Edge scatter: acc[col[e]] += dinv[row[e]]*dinv[col[e]] * H[row[e]]
// 16 threads per edge, 4 channels (float4) each. L2-resident atomics.
// ---------------------------------------------------------------------------
__global__ __launch_bounds__(256) void scatter_kernel(
    const float* __restrict__ H, const int* __restrict__ row,
    const int* __restrict__ col, const float* __restrict__ dinv,
    float* __restrict__ acc, int E)
{
    int gid = blockIdx.x * blockDim.x + threadIdx.x;
    int e   = gid >> 4;
    if (e >= E) return;
    int c4  = (gid & 15) * 4;

    int r = row[e], c = col[e];
    float nrm = dinv[r] * dinv[c];

    const float4 h = *reinterpret_cast<const float4*>(H + (size_t)r * 64 + c4);
    float* dst = acc + (size_t)c * 64 + c4;
    unsafeAtomicAdd(dst + 0, nrm * h.x);
    unsafeAtomicAdd(dst + 1, nrm * h.y);
    unsafeAtomicAdd(dst + 2, nrm * h.z);
    unsafeAtomicAdd(dst + 3, nrm * h.w);
}

// ---------------------------------------------------------------------------
// Graph pooling: sums[batch[n]][c] += h[n][c]; counts[batch[n]] += 1
// ---------------------------------------------------------------------------
__global__ void pool_kernel(const float* __restrict__ H, const int* __restrict__ batch,
                            float* __restrict__ sums, float* __restrict__ counts, int N)
{
    int gid = blockIdx.x * blockDim.x + threadIdx.x;
    int n = gid >> 6;
    if (n >= N) return;
    int c = gid & 63;
    int g = batch[n];
    unsafeAtomicAdd(&sums[g * 64 + c], H[(size_t)n * 64 + c]);
    if (c == 0) unsafeAtomicAdd(&counts[g], 1.0f);
}

// ---------------------------------------------------------------------------
// Final head: out[g] = (sums[g] . Wl) / max(counts) + bl   (one block, 128 thr)
// ---------------------------------------------------------------------------
__global__ __launch_bounds__(128) void final_kernel(
    const float* __restrict__ sums, const float* __restrict__ counts,
    const float* __restrict__ Wl, const float* __restrict__ bl,
    float* __restrict__ out)
{
    __shared__ float nmax_s;
    int g = threadIdx.x;                 // 0..127
    if (g == 0) {
        float m = 0.0f;
        for (int i = 0; i < N_GRAPHS; ++i) m = fmaxf(m, counts[i]);
        nmax_s = m;
    }
    __syncthreads();
    float dot = 0.0f;
    for (int c = 0; c < 64; ++c) dot += sums[g * 64 + c] * Wl[c];
    out[g] = dot / nmax_s + bl[0];
}

// ---------------------------------------------------------------------------
extern "C" void kernel_launch(void* const* d_in, const int* in_sizes, int n_in,
                              void* d_out, int out_size, void* d_ws, size_t ws_size,
                              hipStream_t stream) {
    (void)in_sizes; (void)n_in; (void)out_size; (void)ws_size;

    const float* x     = (const float*)d_in[0];
    const int*   ei    = (const int*)d_in[1];     // [2, E]
    const int*   batch = (const int*)d_in[2];
    const float* W1 = (const float*)d_in[3];  const float* b1 = (const float*)d_in[4];
    const float* W2 = (const float*)d_in[5];  const float* b2 = (const float*)d_in[6];
    const float* W3 = (const float*)d_in[7];  const float* b3 = (const float*)d_in[8];
    const float* W4 = (const float*)d_in[9];  const float* b4 = (const float*)d_in[10];
    const float* Wl = (const float*)d_in[11]; const float* bl = (const float*)d_in[12];

    const int N = N_NODES, E = N_EDGES, G = N_GRAPHS;
    const int* row = ei;
    const int* col = ei + E;

    // Workspace carve-out (~52 MB): hpre | acc | dinv | sums | counts
    float* hpre   = (float*)d_ws;
    float* acc    = hpre + (size_t)N * 64;
    float* dinv   = acc  + (size_t)N * 64;
    float* sums   = dinv + N;
    float* counts = sums + (size_t)G * 64;
    float* out    = (float*)d_out;

    const int TB = 256;

    // --- degree normalization -------------------------------------------------
    zero_kernel  <<<(N + TB - 1) / TB, TB, 0, stream>>>(dinv, N);
    degree_kernel<<<(E + TB - 1) / TB, TB, 0, stream>>>(col, dinv, E);
    dinv_kernel  <<<(N + TB - 1) / TB, TB, 0, stream>>>(dinv, N);

    const int gemm_blocks    = N / 16;                       // 6250, exact
    const int bias_blocks    = (N * 64 + TB - 1) / TB;
    const int scatter_blocks = (E * 16 + TB - 1) / TB;

    // --- layer 1 (Cin = 128) --------------------------------------------------
    gemm_wmma_f32  <<<gemm_blocks, 128, 0, stream>>>(x,   W1, hpre, N, 128);
    init_bias_kernel<<<bias_blocks, TB, 0, stream>>>(acc, b1, N * 64);
    scatter_kernel <<<scatter_blocks, TB, 0, stream>>>(hpre, row, col, dinv, acc, E);

    // --- layers 2..4 (Cin = 64), ping-pong acc -> hpre -> acc -----------------
    gemm_wmma_f32  <<<gemm_blocks, 128, 0, stream>>>(acc, W2, hpre, N, 64);
    init_bias_kernel<<<bias_blocks, TB, 0, stream>>>(acc, b2, N * 64);
    scatter_kernel <<<scatter_blocks, TB, 0, stream>>>(hpre, row, col, dinv, acc, E);

    gemm_wmma_f32  <<<gemm_blocks, 128, 0, stream>>>(acc, W3, hpre, N, 64);
    init_bias_kernel<<<bias_blocks, TB, 0, stream>>>(acc, b3, N * 64);
    scatter_kernel <<<scatter_blocks, TB, 0, stream>>>(hpre, row, col, dinv, acc, E);

    gemm_wmma_f32  <<<gemm_blocks, 128, 0, stream>>>(acc, W4, hpre, N, 64);
    init_bias_kernel<<<bias_blocks, TB, 0, stream>>>(acc, b4, N * 64);
    scatter_kernel <<<scatter_blocks, TB, 0, stream>>>(hpre, row, col, dinv, acc, E);

    // --- pooling + head -------------------------------------------------------
    zero_kernel<<<(G * 64 + G + TB - 1) / TB, TB, 0, stream>>>(sums, G * 64 + G);
    pool_kernel<<<(N * 64 + TB - 1) / TB, TB, 0, stream>>>(acc, batch, sums, counts, N);
    final_kernel<<<1, 128, 0, stream>>>(sums, counts, Wl, bl, out);
}